// Conv2d_72430328481302
// MI455X (gfx1250) — compile-verified
//
#include <hip/hip_runtime.h>

typedef float v2f __attribute__((ext_vector_type(2)));
typedef float v8f __attribute__((ext_vector_type(8)));

#define C_IN   128
#define K_OUT  128
#define W_     56
#define HW_    3136          // 56*56
#define KRED   1152          // C_IN * 9

// grid: (392, 8); block: 256 threads = 8 wave32s.
// blockIdx.y selects a 16-output-channel tile; blockIdx.x selects 128
// consecutive output spatial positions (each wave owns 16 of them).
// Implicit-GEMM: D(16oc x 16sp) += A(16oc x 4k) * B(4k x 16sp) via
// V_WMMA_F32_16X16X4_F32, k = (r,s)-major, channel-minor.
__global__ __launch_bounds__(256)
void conv3x3_wmma_f32(const float* __restrict__ in,
                      const float* __restrict__ wgt,
                      const float* __restrict__ bias,
                      float* __restrict__ out)
{
    // Weights staged as wlds[k*16 + oc_local], k = rs*128 + c  (72 KB)
    __shared__ float wlds[16 * KRED];

    const int tid     = threadIdx.x;
    const int oc_base = blockIdx.y * 16;

    // ---- stage weight tile: global (oc,c,r,s) -> LDS (rs,c,oc) ----
    {
        const float* wsrc = wgt + (long)oc_base * KRED;
        for (int g = tid; g < 16 * KRED; g += 256) {
            int oc_l = g / KRED;
            int rem  = g - oc_l * KRED;   // c*9 + rs
            int c    = rem / 9;
            int rs   = rem - c * 9;
            wlds[(rs * C_IN + c) * 16 + oc_l] = wsrc[g];  // coalesced read
        }
    }
    __syncthreads();

    const int wave = tid >> 5;
    const int lane = tid & 31;
    const int m    = lane & 15;          // A: M row;  B/D: N column
    const int koff = (lane >> 4) << 1;   // lanes 16-31 hold K+2, K+3

    // per-lane output spatial position
    const int p  = blockIdx.x * 128 + wave * 16 + m;
    const int n  = p / HW_;
    const int hw = p - n * HW_;
    const int h  = hw / W_;
    const int w  = hw - h * W_;

    v8f acc = {};

    #pragma unroll
    for (int rs = 0; rs < 9; ++rs) {
        const int r  = rs / 3;
        const int s  = rs - r * 3;
        const int ih = h + r - 1;
        const int iw = w + s - 1;
        const bool valid = (ih >= 0) & (ih < W_) & (iw >= 0) & (iw < W_);
        const float mask = valid ? 1.0f : 0.0f;          // zero padding
        const long  off  = valid ? ((long)n * C_IN * HW_ + (long)ih * W_ + iw)
                                 : 0;                    // clamped, masked
        const float* gin = in + off + (long)koff * HW_;
        const float* wl  = &wlds[(rs * C_IN + koff) * 16 + m];

        #pragma unroll 4
        for (int c = 0; c < C_IN; c += 4) {
            v2f A, B;
            A.x = wl[(c    ) * 16];                      // A[M=m][k0]
            A.y = wl[(c + 1) * 16];                      // A[M=m][k0+1]
            B.x = gin[(long)(c    ) * HW_] * mask;       // B[k0  ][N=m]
            B.y = gin[(long)(c + 1) * HW_] * mask;       // B[k0+1][N=m]
            acc = __builtin_amdgcn_wmma_f32_16x16x4_f32(
                false, A, false, B, (short)0, acc, false, false);
        }
    }

    // ---- bias + store: VGPR g holds M = m_hi + g, N = m ----
    const int m_hi = (lane >> 4) << 3;   // 0 or 8
    #pragma unroll
    for (int g = 0; g < 8; ++g) {
        const int oc = oc_base + m_hi + g;
        out[((long)n * K_OUT + oc) * HW_ + hw] = acc[g] + bias[oc];
    }
}

extern "C" void kernel_launch(void* const* d_in, const int* in_sizes, int n_in,
                              void* d_out, int out_size, void* d_ws, size_t ws_size,
                              hipStream_t stream) {
    (void)in_sizes; (void)n_in; (void)d_ws; (void)ws_size; (void)out_size;
    const float* in   = (const float*)d_in[0];   // (16,128,56,56) f32
    const float* wgt  = (const float*)d_in[1];   // (128,128,3,3)  f32
    const float* bias = (const float*)d_in[2];   // (128,)         f32
    float*       out  = (float*)d_out;           // (16,128,56,56) f32

    dim3 grid(392, 8, 1);   // 392 spatial tiles of 128 positions, 8 oc tiles
    dim3 block(256, 1, 1);  // 8 wave32s
    conv3x3_wmma_f32<<<grid, block, 0, stream>>>(in, wgt, bias, out);
}